// EMA_74328704025070
// MI455X (gfx1250) — compile-verified
//
#include <hip/hip_runtime.h>
#include <hip/hip_bf16.h>
#include <math.h>

// Problem constants (from reference): x[B, P, F]
#define EMA_B 8
#define EMA_P 4096
#define EMA_F 1024
#define EMA_C 16          // chunks along P
#define EMA_L 256         // P / C, time steps per chunk

typedef float v2f __attribute__((ext_vector_type(2)));
typedef float v8f __attribute__((ext_vector_type(8)));

__device__ __forceinline__ float sigmoidf_(float v) {
    return 1.0f / (1.0f + __expf(-v));
}

// small integer power (n <= 15 in practice)
__device__ __forceinline__ float powi_f(float a, int n) {
    float r = 1.0f;
    for (int i = 0; i < n; ++i) r *= a;
    return r;
}

// ---------------------------------------------------------------------------
// Pass A: per (b, f, chunk) compute chunk-local EMA end value (zero initial h).
// Reads all of x once (warms the 192MB L2), writes B*C*F floats to workspace.
// ---------------------------------------------------------------------------
__global__ void ema_chunk_ends(const float* __restrict__ x,
                               const float* __restrict__ raw_alpha,
                               float* __restrict__ ws) {
    int g = blockIdx.x * blockDim.x + threadIdx.x;      // [0, B*F*C)
    int f = g & (EMA_F - 1);                            // consecutive threads -> consecutive f
    int j = (g >> 10) & (EMA_C - 1);                    // chunk
    int b = g >> 14;

    float a  = sigmoidf_(raw_alpha[0]);
    float om = 1.0f - a;

    const float* xp = x + ((size_t)b * EMA_P + (size_t)j * EMA_L) * EMA_F + f;

    float h = 0.0f;
#pragma unroll 8
    for (int t = 0; t < EMA_L; ++t) {
        h = fmaf(a, h, om * xp[(size_t)t * EMA_F]);     // regular (RT) load: keep in L2
    }
    ws[((size_t)b * EMA_C + j) * EMA_F + f] = h;
}

// ---------------------------------------------------------------------------
// Pass B: one wave per (b, f-tile-of-16, chunk).
// Carry c_j combined from chunk-end values, then 16-step blocks computed as
// H = L(16x16) * X(16x16) + alpha^{t+1} (x) c   via 4x V_WMMA_F32_16X16X4_F32.
// ---------------------------------------------------------------------------
__global__ void ema_scan_wmma(const float* __restrict__ x,
                              const float* __restrict__ raw_alpha,
                              const float* __restrict__ ws,
                              float* __restrict__ out) {
    const int lane = threadIdx.x & 31;
    const int wv   = blockIdx.x * (blockDim.x >> 5) + (threadIdx.x >> 5);
    // wv in [0, B * (F/16) * C)
    const int ftile = wv & 63;                // F/16 = 64 tiles
    const int j     = (wv >> 6) & (EMA_C - 1);
    const int b     = wv >> 10;

    const int nlo  = lane & 15;               // N (column / f within tile)
    const int half = lane >> 4;               // lane half selects K/M offsets
    const int f    = (ftile << 4) + nlo;

    const float a  = sigmoidf_(raw_alpha[0]);
    const float om = 1.0f - a;

    // alpha^L (L=256) by repeated squaring
    float aL = a;
#pragma unroll
    for (int i = 0; i < 8; ++i) aL *= aL;

    // Carry into chunk j: serial combine of previous chunk end values (<=15 steps)
    float c = 0.0f;
    for (int i = 0; i < j; ++i) {
        c = fmaf(aL, c, ws[((size_t)b * EMA_C + i) * EMA_F + f]);
    }

    // A-matrix (lower-triangular decay), 4 K-slices of 16x4.
    // A layout: lanes0-15 M=0..15; VGPR0 holds K=k0 (lo half) / K=k0+2 (hi half), VGPR1 K=k0+1 / k0+3.
    v2f Am[4];
#pragma unroll
    for (int s = 0; s < 4; ++s) {
        int m  = nlo;
        int k0 = 4 * s + 2 * half;
        Am[s].x = (k0     <= m) ? om * powi_f(a, m - k0)     : 0.0f;
        Am[s].y = (k0 + 1 <= m) ? om * powi_f(a, m - k0 - 1) : 0.0f;
    }

    // Carry row multipliers: accumulator VGPR r holds row M = r + 8*half -> alpha^{M+1}
    float mrow[8];
    mrow[0] = powi_f(a, half ? 9 : 1);
#pragma unroll
    for (int r = 1; r < 8; ++r) mrow[r] = mrow[r - 1] * a;

    const float* xp = x   + ((size_t)b * EMA_P + (size_t)j * EMA_L) * EMA_F + f;
    float*       op = out + ((size_t)b * EMA_P + (size_t)j * EMA_L) * EMA_F + f;

    // Double-buffered X (B-matrix) loads. B layout mirrors A with N across lanes:
    // slice s, VGPR0 = row K=4s+2*half, VGPR1 = row K=4s+2*half+1.
    v2f Bv[2][4];
#pragma unroll
    for (int s = 0; s < 4; ++s) {
        int t = 4 * s + 2 * half;
        Bv[0][s].x = __builtin_nontemporal_load(xp + (size_t)t       * EMA_F);
        Bv[0][s].y = __builtin_nontemporal_load(xp + (size_t)(t + 1) * EMA_F);
    }

#pragma unroll
    for (int blk = 0; blk < EMA_L / 16; ++blk) {
        const int cur = blk & 1;
        if (blk < (EMA_L / 16 - 1)) {
            const int nt0 = (blk + 1) * 16;
#pragma unroll
            for (int s = 0; s < 4; ++s) {
                int t = nt0 + 4 * s + 2 * half;
                Bv[cur ^ 1][s].x = __builtin_nontemporal_load(xp + (size_t)t       * EMA_F);
                Bv[cur ^ 1][s].y = __builtin_nontemporal_load(xp + (size_t)(t + 1) * EMA_F);
            }
        }

        // accumulator preloaded with carry contribution alpha^{t+1} * c
        v8f acc;
#pragma unroll
        for (int r = 0; r < 8; ++r) acc[r] = mrow[r] * c;

#pragma unroll
        for (int s = 0; s < 4; ++s) {
            acc = __builtin_amdgcn_wmma_f32_16x16x4_f32(
                false, Am[s], false, Bv[cur][s], (short)0, acc, false, false);
        }

        // store 16x16 result block; NT so out stream doesn't evict x from L2
        const int tbase = blk * 16 + half * 8;
#pragma unroll
        for (int r = 0; r < 8; ++r) {
            __builtin_nontemporal_store(acc[r], op + (size_t)(tbase + r) * EMA_F);
        }

        // next carry = H[15, f]: held in acc[7] of lanes 16..31 (M=15, N=lane-16)
        c = __shfl(acc[7], 16 + nlo, 32);
    }
}

extern "C" void kernel_launch(void* const* d_in, const int* in_sizes, int n_in,
                              void* d_out, int out_size, void* d_ws, size_t ws_size,
                              hipStream_t stream) {
    const float* x  = (const float*)d_in[0];
    const float* ra = (const float*)d_in[1];
    float* out = (float*)d_out;
    float* ws  = (float*)d_ws;   // needs B*C*F*4 = 512 KB

    // Pass A: chunk-local end values (one thread per (b,f,chunk))
    {
        const int threads = EMA_B * EMA_F * EMA_C;    // 131072
        ema_chunk_ends<<<threads / 256, 256, 0, stream>>>(x, ra, ws);
    }
    // Pass B: WMMA blocked scan (one wave per (b, f-tile, chunk))
    {
        const int waves = EMA_B * (EMA_F / 16) * EMA_C;   // 8192 waves
        ema_scan_wmma<<<waves / 8, 256, 0, stream>>>(x, ra, ws, out);
    }
}